// ApplyPolicyMap_42683384987871
// MI455X (gfx1250) — compile-verified
//
#include <hip/hip_runtime.h>
#include <stdint.h>

#define IN_DIM  5120   // 80 planes * 64 squares
#define OUT_DIM 1858   // lc0 move encoding

// -------- CDNA5 async global->LDS path (gfx1250), guarded so compile never breaks
#if defined(__HIP_DEVICE_COMPILE__) &&                                        \
    __has_builtin(__builtin_amdgcn_global_load_async_to_lds_b128) &&          \
    __has_builtin(__builtin_amdgcn_s_wait_asynccnt)
#define HAS_ASYNC_LDS 1
#else
#define HAS_ASYNC_LDS 0
#endif

// Builtin signature (from hipcc diagnostic): arg0 = int4-vector pointer in
// AS1 (global, printed as __device__), arg1 = LDS-side pointer (AS3), then
// two immediate ints (offset, cpol).
typedef int v4i __attribute__((ext_vector_type(4)));
typedef __attribute__((address_space(1))) v4i gv4i;  // global int4
typedef __attribute__((address_space(3))) v4i lv4i;  // LDS int4

// ---------------------------------------------------------------------------
// Kernel 1: recover the one-hot row index of each fc1 column.
// fc1 is [IN_DIM, OUT_DIM] row-major; consecutive lanes read consecutive
// columns -> fully coalesced. Each column has exactly one non-zero, so exactly
// one (block.y, thread) pair writes idx[j] -> deterministic every call.
// ---------------------------------------------------------------------------
__global__ __launch_bounds__(256) void build_index_kernel(
    const float* __restrict__ fc1, int* __restrict__ idx) {
  const int j = blockIdx.x * 256 + threadIdx.x;
  if (j >= OUT_DIM) return;
  const int i0 = blockIdx.y * 64;
#pragma unroll 4
  for (int r = 0; r < 64; ++r) {
    const int i = i0 + r;
    if (fc1[(size_t)i * OUT_DIM + j] != 0.0f) idx[j] = i;
  }
}

// ---------------------------------------------------------------------------
// Kernel 2: one block per batch row. Stage the 20KB x-row into LDS with the
// gfx1250 async global->LDS DMA path (ASYNCcnt-tracked), then gather the 1858
// outputs from LDS (ds_load_b32) with coalesced stores.
// ---------------------------------------------------------------------------
__global__ __launch_bounds__(256) void policy_gather_kernel(
    const float* __restrict__ x, const int* __restrict__ idx,
    float* __restrict__ out) {
  __shared__ float row[IN_DIM];  // 20480 bytes
  const int b = blockIdx.x;
  const unsigned t = threadIdx.x;
  const float* src = x + (size_t)b * IN_DIM;

#if HAS_ASYNC_LDS
  // 1280 lanes-worth of 16B async copies: 5 per thread. Per-lane global VGPR
  // address -> per-lane LDS VGPR address (GLOBAL_LOAD_ASYNC_TO_LDS_B128).
#pragma unroll
  for (int c = 0; c < 5; ++c) {
    const unsigned off = (unsigned)c * 4096u + t * 16u;
    const char* g = (const char*)src + off;
    char*       l = (char*)row + off;
    __builtin_amdgcn_global_load_async_to_lds_b128(
        (gv4i*)g, (lv4i*)l, /*imm offset*/ 0, /*cpol*/ 0);
  }
  // Wait for this wave's async copies, then barrier so all waves' LDS tiles
  // are visible (ASYNCcnt semantics per CDNA5 ISA).
  __builtin_amdgcn_s_wait_asynccnt(0);
  __syncthreads();
#else
  // Fallback: coalesced float4 loads + ds stores.
  const float4* src4 = (const float4*)src;
  float4* row4 = (float4*)row;
#pragma unroll
  for (int c = 0; c < 5; ++c) row4[t + (unsigned)c * 256u] = src4[t + (unsigned)c * 256u];
  __syncthreads();
#endif

  float* dst = out + (size_t)b * OUT_DIM;
#pragma unroll
  for (int j = (int)t; j < OUT_DIM; j += 256) {
    dst[j] = row[idx[j]];  // idx hot in L2; row gather hits LDS banks
  }
}

// ---------------------------------------------------------------------------
extern "C" void kernel_launch(void* const* d_in, const int* in_sizes, int n_in,
                              void* d_out, int out_size, void* d_ws, size_t ws_size,
                              hipStream_t stream) {
  const float* x   = (const float*)d_in[0];  // [B, 80, 8, 8] fp32
  const float* fc1 = (const float*)d_in[1];  // [5120, 1858] fp32 one-hot cols
  float* out = (float*)d_out;                // [B, 1858] fp32
  int* idx = (int*)d_ws;                     // 1858 ints of scratch

  const int B = in_sizes[0] / IN_DIM;

  dim3 g1((OUT_DIM + 255) / 256, IN_DIM / 64);  // (8, 80)
  build_index_kernel<<<g1, 256, 0, stream>>>(fc1, idx);

  policy_gather_kernel<<<B, 256, 0, stream>>>(x, idx, out);
}